// TopoLayer_73813307949741
// MI455X (gfx1250) — compile-verified
//
#include <hip/hip_runtime.h>
#include <stdint.h>
#include <math.h>

// Problem constants (reference shapes are fixed).
#define IMG_H 256
#define IMG_W 256
#define GRES  256            // filtration grid resolution per parameter
#define G2    (GRES * GRES)  // 65536 bins

// LDS layout for the histogram kernel (dynamic shared):
//   [0,            262144)  int32 hist[G2]                 (256 KB)
//   [262144,       266240)  float pixbuf[2][IMG_W*2]       (2 x 2 KB row staging)
//   [266240,       268288)  u32   idxbuf[2][IMG_W]         (2 x 1 KB packed indices)
#define HIST_BYTES   (G2 * 4)
#define PIXROW_BYTES (IMG_W * 2 * 4)
#define PIXBUF_OFF   HIST_BYTES
#define IDXROW_BYTES (IMG_W * 4)
#define IDXBUF_OFF   (PIXBUF_OFF + 2 * PIXROW_BYTES)
#define SHMEM_BYTES  (IDXBUF_OFF + 2 * IDXROW_BYTES)   // 268288 < 320 KB/WGP

// grid[i] = lo + (hi-lo) * (i/255), mirroring jnp.linspace(0,1,256) scaled.
__device__ __forceinline__ float gridv(float lo, float range, int i) {
    return __fadd_rn(lo, __fmul_rn(range, __fmul_rn((float)i, 1.0f / 255.0f)));
}

// searchsorted-left over gridv(0..255): smallest i in [0,256] with grid[i] >= f.
__device__ __forceinline__ int ssidx(float f, float lo, float range, float inv) {
    if (f == __builtin_inff()) return GRES;       // padded boundary pixels
    float t = (f - lo) * inv;                     // ~ 255*(f-lo)/range
    t = fminf(fmaxf(t, 0.0f), 256.0f);
    int i = (int)ceilf(t);
    while (i > 0    && gridv(lo, range, i - 1) >= f) --i;
    while (i < GRES && gridv(lo, range, i)     <  f) ++i;
    return i;
}

// gfx1250 async global->LDS copy (ASYNCcnt-tracked); 8 bytes per lane.
__device__ __forceinline__ void async_ld_b64(uint32_t lds_off, unsigned long long gaddr) {
    asm volatile("global_load_async_to_lds_b64 %0, %1, off"
                 :: "v"(lds_off), "v"(gaddr) : "memory");
}
__device__ __forceinline__ void wait_async0() {
    asm volatile("s_wait_asynccnt 0x0" ::: "memory");
}

// ---------------------------------------------------------------------------
// Kernel 1: per-image channelwise min/max. One block per image.
// ws layout per image: [lo0, hi0, lo1, hi1]
// ---------------------------------------------------------------------------
__global__ __launch_bounds__(256) void esm_minmax_kernel(const float* __restrict__ in,
                                                         float* __restrict__ ws) {
    const int n   = blockIdx.x;
    const int tid = threadIdx.x;
    const float4* img4 = (const float4*)(in + (size_t)n * (IMG_H * IMG_W * 2));
    const int nvec = IMG_H * IMG_W * 2 / 4;       // 32768 float4 (2 pixels each)

    float mn0 =  __builtin_inff(), mx0 = -__builtin_inff();
    float mn1 =  __builtin_inff(), mx1 = -__builtin_inff();
    for (int i = tid; i < nvec; i += blockDim.x) {
        __builtin_prefetch(&img4[i + blockDim.x], 0, 1);   // global_prefetch_b8
        float4 v = img4[i];                                 // (c0,c1,c0,c1)
        mn0 = fminf(mn0, fminf(v.x, v.z));
        mx0 = fmaxf(mx0, fmaxf(v.x, v.z));
        mn1 = fminf(mn1, fminf(v.y, v.w));
        mx1 = fmaxf(mx1, fmaxf(v.y, v.w));
    }

    __shared__ float smn0[256], smx0[256], smn1[256], smx1[256];
    smn0[tid] = mn0; smx0[tid] = mx0; smn1[tid] = mn1; smx1[tid] = mx1;
    __syncthreads();
    for (int s = 128; s > 0; s >>= 1) {
        if (tid < s) {
            smn0[tid] = fminf(smn0[tid], smn0[tid + s]);
            smx0[tid] = fmaxf(smx0[tid], smx0[tid + s]);
            smn1[tid] = fminf(smn1[tid], smn1[tid + s]);
            smx1[tid] = fmaxf(smx1[tid], smx1[tid + s]);
        }
        __syncthreads();
    }
    if (tid == 0) {
        ws[n * 4 + 0] = smn0[0];
        ws[n * 4 + 1] = smx0[0];
        ws[n * 4 + 2] = smn1[0];
        ws[n * 4 + 3] = smx1[0];
    }
}

// ---------------------------------------------------------------------------
// Kernel 2: Euler signed-measure histogram, one workgroup per image.
// Full 256x256 int32 histogram in LDS (CDNA5: 320 KB per WGP); pixel rows are
// streamed in with async global->LDS b64 copies and indexed exactly once.
// ---------------------------------------------------------------------------
extern __shared__ unsigned char lds_raw[];

__global__ __launch_bounds__(256) void esm_hist_kernel(const float* __restrict__ in,
                                                       const float* __restrict__ ws,
                                                       float* __restrict__ out) {
    const int n   = blockIdx.x;
    const int tid = threadIdx.x;

    int* hist = (int*)lds_raw;
    const float* img = in + (size_t)n * (IMG_H * IMG_W * 2);

    const float lo0 = ws[n * 4 + 0], hi0 = ws[n * 4 + 1];
    const float lo1 = ws[n * 4 + 2], hi1 = ws[n * 4 + 3];
    const float r0 = hi0 - lo0, r1 = hi1 - lo1;
    const float inv0 = (r0 > 0.0f) ? (255.0f / r0) : 0.0f;
    const float inv1 = (r1 > 0.0f) ? (255.0f / r1) : 0.0f;

    // Async-stage pixel row pr (2 KB) into pixbuf[pr&1]: 256 lanes x 8 bytes.
    auto stage = [&](int pr) {
        uint32_t lds_off = (uint32_t)(uintptr_t)
            (lds_raw + PIXBUF_OFF + (pr & 1) * PIXROW_BYTES + tid * 8);
        unsigned long long ga = (unsigned long long)(uintptr_t)
            (img + (size_t)pr * (IMG_W * 2) + tid * 2);
        async_ld_b64(lds_off, ga);
    };
    // Compute packed grid indices (i1<<16 | i0) for pixel row pr, once per pixel.
    auto idxcalc = [&](int pr) {
        const float2* prow = (const float2*)(lds_raw + PIXBUF_OFF + (pr & 1) * PIXROW_BYTES);
        uint32_t* irow = (uint32_t*)(lds_raw + IDXBUF_OFF + (pr & 1) * IDXROW_BYTES);
        float2 v = prow[tid];
        uint32_t i0 = (uint32_t)ssidx(v.x, lo0, r0, inv0);
        uint32_t i1 = (uint32_t)ssidx(v.y, lo1, r1, inv1);
        irow[tid] = (i1 << 16) | i0;
    };

    // Overlap: start streaming pixel row 0 while zeroing the histogram.
    stage(0);
    for (int k = tid * 4; k < G2; k += 256 * 4)
        *(int4*)&hist[k] = make_int4(0, 0, 0, 0);   // ds_store_b128
    wait_async0();
    __syncthreads();
    idxcalc(0);
    __syncthreads();

    const uint32_t OOBP = (256u << 16) | 256u;      // +inf padding -> index 256
    auto pmin = [](uint32_t a, uint32_t b) {
        uint32_t l = (a & 0xFFFFu) < (b & 0xFFFFu) ? (a & 0xFFFFu) : (b & 0xFFFFu);
        uint32_t h = (a >> 16)     < (b >> 16)     ? (a >> 16)     : (b >> 16);
        return (h << 16) | l;
    };
    auto dep = [&](uint32_t p, int w) {
        int flat = (int)(p & 0xFFFFu) * GRES + (int)(p >> 16);
        if (flat < G2) atomicAdd(&hist[flat], w);   // ds_add_u32; JAX drops OOB
    };

    for (int r = 0; r <= IMG_H; ++r) {              // 257 vertex rows
        const bool more = (r < IMG_H - 1);          // pixel rows 1..255 pending
        if (more) stage(r + 1);                     // overlap copy with deposits

        const uint32_t* up = (const uint32_t*)(lds_raw + IDXBUF_OFF + ((r - 1) & 1) * IDXROW_BYTES);
        const uint32_t* dn = (const uint32_t*)(lds_raw + IDXBUF_OFF + (r & 1) * IDXROW_BYTES);
        const bool rm = (r > 0), rp = (r < IMG_H);

        for (int c = tid; c <= IMG_W; c += 256) {   // 257 vertex columns
            const bool cm = (c > 0), cp = (c < IMG_W);
            uint32_t A = (rm && cm) ? up[c - 1] : OOBP;   // x[r-1,c-1]
            uint32_t B = (rm && cp) ? up[c]     : OOBP;   // x[r-1,c]
            uint32_t C = (rp && cm) ? dn[c - 1] : OOBP;   // x[r,  c-1]
            uint32_t D = (rp && cp) ? dn[c]     : OOBP;   // x[r,  c]

            dep(pmin(pmin(A, B), pmin(C, D)), +1);        // vertex (0-cell)
            if (cp)       dep(pmin(B, D), -1);            // horizontal edge
            if (rp)       dep(pmin(C, D), -1);            // vertical edge
            if (rp && cp) dep(D, +1);                     // pixel (2-cell)
        }
        __syncthreads();

        if (more) {
            wait_async0();      // drain this wave's async copies of row r+1
            __syncthreads();    // all waves' copies visible in LDS
            idxcalc(r + 1);
            __syncthreads();
        }
    }

    // Formatted output: rows (grid_x[i], grid_y[j], hist[i*G+j]); fully
    // coalesced float4 stores of the flat [G*G*3] stream.
    float* o = out + (size_t)n * (G2 * 3);
    for (int q = tid * 4; q < G2 * 3; q += 256 * 4) {
        float4 v;
#pragma unroll
        for (int e = 0; e < 4; ++e) {
            const int f    = q + e;
            const int k    = (int)((unsigned)f / 3u);
            const int comp = f - k * 3;
            const int i = k >> 8, j = k & (GRES - 1);
            float val;
            if      (comp == 0) val = gridv(lo0, r0, i);
            else if (comp == 1) val = gridv(lo1, r1, j);
            else                val = (float)hist[k];     // ds_load_b32
            ((float*)&v)[e] = val;
        }
        *(float4*)&o[q] = v;   // global_store_b128
    }
}

// ---------------------------------------------------------------------------
extern "C" void kernel_launch(void* const* d_in, const int* in_sizes, int n_in,
                              void* d_out, int out_size, void* d_ws, size_t ws_size,
                              hipStream_t stream) {
    const float* in  = (const float*)d_in[0];
    float*       out = (float*)d_out;
    float*       ws  = (float*)d_ws;                   // N*4 floats of lo/hi
    const int N = in_sizes[0] / (IMG_H * IMG_W * 2);   // 128 images

    esm_minmax_kernel<<<N, 256, 0, stream>>>(in, ws);

    (void)hipFuncSetAttribute((const void*)esm_hist_kernel,
                              hipFuncAttributeMaxDynamicSharedMemorySize,
                              (int)SHMEM_BYTES);
    esm_hist_kernel<<<N, 256, SHMEM_BYTES, stream>>>(in, ws, out);
}